// CLRNetAssign_17076789969268
// MI455X (gfx1250) — compile-verified
//
#include <hip/hip_runtime.h>
#include <hip/hip_bf16.h>
#include <cstdint>

#define B_ 4
#define N_ 4096
#define T_ 32
#define S_ 72
#define P_ 78
#define Q_ 4
#define TPB 256
#define NTILES (N_ / TPB)   // 16

// ---------------------------------------------------------------------------
// CDNA5 async global->LDS staging (gfx1250 path). The clang-22 builtin takes
// (v4i AS1* gsrc, v4i AS3* lds_dst, imm offset, imm cpol) where v4i is a GCC
// vector 'int __attribute__((vector_size(16)))' (param printed as
// "...vector_size(4*sizeof(int)) int __device__ *" in the round-1 diagnostic).
// ---------------------------------------------------------------------------
#if defined(__HIP_DEVICE_COMPILE__) && __has_builtin(__builtin_amdgcn_global_load_async_to_lds_b128)
#define USE_ASYNC_LDS 1
#else
#define USE_ASYNC_LDS 0
#endif

typedef int v4i_gcc __attribute__((vector_size(16)));

__device__ __forceinline__ void stage_tile_vec4(const float* __restrict__ g,
                                                float* __restrict__ s, int nvec4) {
#if USE_ASYNC_LDS
  for (int i = threadIdx.x; i < nvec4; i += (int)blockDim.x) {
    __builtin_amdgcn_global_load_async_to_lds_b128(
        (__attribute__((address_space(1))) v4i_gcc*)(uintptr_t)(g + 4 * i),
        (__attribute__((address_space(3))) v4i_gcc*)(uint32_t)(uintptr_t)(s + 4 * i),
        0, 0);
  }
#if __has_builtin(__builtin_amdgcn_s_wait_asynccnt)
  __builtin_amdgcn_s_wait_asynccnt(0);
#else
  asm volatile("s_wait_asynccnt 0" ::: "memory");
#endif
#else
  for (int i = threadIdx.x; i < nvec4; i += (int)blockDim.x) {
    reinterpret_cast<float4*>(s)[i] = reinterpret_cast<const float4*>(g)[i];
  }
#endif
}

// ---------------------------------------------------------------------------
// sorted-descending top-4 insert with JAX top_k tie-breaking:
// higher value wins; equal value -> lower index wins.
// ---------------------------------------------------------------------------
__device__ __forceinline__ void insert4(float* v, int* ix, float nv, int ni) {
#pragma unroll
  for (int q = 0; q < Q_; ++q) {
    bool better = (nv > v[q]) || ((nv == v[q]) && (ni < ix[q]));
    if (better) {
#pragma unroll
      for (int r = Q_ - 1; r > q; --r) { v[r] = v[r - 1]; ix[r] = ix[r - 1]; }
      v[q] = nv; ix[q] = ni;
      return;
    }
  }
}

// ---------------------------------------------------------------------------
__global__ void k_init(unsigned int* ws_max) {
  if (threadIdx.x < B_ * 3) ws_max[threadIdx.x] = 0u;
}

// phase1: iou + per-batch masked maxima of (distances, sd, td)
__global__ __launch_bounds__(TPB) void k_phase1(
    const float* __restrict__ preds, const float* __restrict__ tgts,
    const int* __restrict__ masks, const int* __restrict__ iwp,
    float* __restrict__ ws_iou, unsigned int* __restrict__ ws_max) {
  __shared__ float s_tgt[T_ * P_];
  __shared__ int   s_mask[T_];
  const int b = blockIdx.x / NTILES;
  const int n = (blockIdx.x % NTILES) * TPB + threadIdx.x;

  stage_tile_vec4(tgts + (size_t)b * T_ * P_, s_tgt, (T_ * P_) / 4);
  if (threadIdx.x < T_) s_mask[threadIdx.x] = masks[b * T_ + threadIdx.x];
  __syncthreads();

  const float iw = (float)iwp[0];
  const float* pr = preds + ((size_t)(b * N_ + n)) * P_;
  __builtin_prefetch(pr, 0, 3);

  float px[S_];
#pragma unroll
  for (int s = 0; s < S_; ++s) px[s] = pr[6 + s];
  const float psx = pr[2], psy = pr[3], pth = pr[4];

  float dmax = 0.f, smax = 0.f, tmax = 0.f;
  for (int t = 0; t < T_; ++t) {
    const float* tr = &s_tgt[t * P_];
    float dsum = 0.f, ovr = 0.f, uni = 0.f;
    int cnt = 0;
#pragma unroll
    for (int s = 0; s < S_; ++s) {
      const float tv = tr[6 + s];
      if ((tv >= 0.f) && (tv < iw)) {
        const float a = px[s];
        dsum += fabsf(a - tv);
        ovr  += fminf(a + 15.f, tv + 15.f) - fmaxf(a - 15.f, tv - 15.f);
        uni  += fmaxf(a + 15.f, tv + 15.f) - fminf(a - 15.f, tv - 15.f);
        cnt++;
      }
    }
    const float vlen = (float)(cnt > 0 ? cnt : 1);
    const float dist = dsum / (vlen + 1e-6f);
    const float iou  = fmaxf(ovr / (uni + 1e-9f), 0.f);
    const float dx = psx - tr[2], dy = psy - tr[3];
    const float sd = sqrtf(dx * dx + dy * dy);
    const float td = fabsf(pth - tr[4]);
    if (s_mask[t]) {
      dmax = fmaxf(dmax, dist);
      smax = fmaxf(smax, sd);
      tmax = fmaxf(tmax, td);
    }
    ws_iou[((size_t)(b * N_ + n)) * T_ + t] = iou;
  }
  atomicMax(&ws_max[b * 3 + 0], __float_as_uint(dmax));
  atomicMax(&ws_max[b * 3 + 1], __float_as_uint(smax));
  atomicMax(&ws_max[b * 3 + 2], __float_as_uint(tmax));
}

// phase2: full cost matrix
__global__ __launch_bounds__(TPB) void k_phase2(
    const float* __restrict__ preds, const float* __restrict__ tgts,
    const int* __restrict__ masks, const int* __restrict__ iwp,
    float* __restrict__ ws_cost, const unsigned int* __restrict__ ws_max) {
  __shared__ float s_tgt[T_ * P_];
  __shared__ int   s_mask[T_];
  const int b = blockIdx.x / NTILES;
  const int n = (blockIdx.x % NTILES) * TPB + threadIdx.x;

  stage_tile_vec4(tgts + (size_t)b * T_ * P_, s_tgt, (T_ * P_) / 4);
  if (threadIdx.x < T_) s_mask[threadIdx.x] = masks[b * T_ + threadIdx.x];
  __syncthreads();

  const float iw = (float)iwp[0];
  const float dmax = fmaxf(__uint_as_float(ws_max[b * 3 + 0]), 1e-6f);
  const float smax = fmaxf(__uint_as_float(ws_max[b * 3 + 1]), 1e-6f);
  const float tmax = fmaxf(__uint_as_float(ws_max[b * 3 + 2]), 1e-6f);

  const float* pr = preds + ((size_t)(b * N_ + n)) * P_;
  __builtin_prefetch(pr, 0, 3);

  float px[S_];
#pragma unroll
  for (int s = 0; s < S_; ++s) px[s] = pr[6 + s];
  const float psx = pr[2], psy = pr[3], pth = pr[4];

  // focal-style class cost per pred, both classes
  const float p0 = 1.f / (1.f + expf(-pr[0]));
  const float p1 = 1.f / (1.f + expf(-pr[1]));
  const float pos0 = -logf(p0 + 1e-12f) * 0.25f * (1.f - p0) * (1.f - p0);
  const float neg0 = -logf(1.f - p0 + 1e-12f) * 0.75f * p0 * p0;
  const float pos1 = -logf(p1 + 1e-12f) * 0.25f * (1.f - p1) * (1.f - p1);
  const float neg1 = -logf(1.f - p1 + 1e-12f) * 0.75f * p1 * p1;
  const float c0 = pos0 - neg0, c1 = pos1 - neg1;

  for (int t = 0; t < T_; ++t) {
    const size_t oidx = ((size_t)(b * N_ + n)) * T_ + t;
    if (!s_mask[t]) { ws_cost[oidx] = __builtin_inff(); continue; }
    const float* tr = &s_tgt[t * P_];
    float dsum = 0.f;
    int cnt = 0;
#pragma unroll
    for (int s = 0; s < S_; ++s) {
      const float tv = tr[6 + s];
      if ((tv >= 0.f) && (tv < iw)) { dsum += fabsf(px[s] - tv); cnt++; }
    }
    const float vlen = (float)(cnt > 0 ? cnt : 1);
    const float dist = dsum / (vlen + 1e-6f);
    const float dx = psx - tr[2], dy = psy - tr[3];
    const float sd = sqrtf(dx * dx + dy * dy);
    const float td = fabsf(pth - tr[4]);
    const float ds = (1.f - dist / dmax) + 0.01f;
    const float ss = (1.f - sd / smax) + 0.01f;
    const float ts = (1.f - td / tmax) + 0.01f;
    const int lab = (int)tr[1];
    const float prod = (ds * ss) * ts;
    ws_cost[oidx] = -(prod * prod) * 3.f + (lab ? c1 : c0);
  }
}

// topk: per (b,t) -> dyn_k and top-4 lowest-cost pred indices
__global__ __launch_bounds__(TPB) void k_topk(
    const float* __restrict__ ws_iou, const float* __restrict__ ws_cost,
    const int* __restrict__ masks, int* __restrict__ ws_dynk,
    int* __restrict__ ws_topk) {
  __shared__ float sv[TPB * Q_];
  __shared__ int   si[TPB * Q_];
  const int b = blockIdx.x / T_;
  const int t = blockIdx.x % T_;

  float iv[Q_], cv[Q_];
  int ii[Q_], ci[Q_];
#pragma unroll
  for (int q = 0; q < Q_; ++q) {
    iv[q] = -__builtin_inff(); ii[q] = 0x7fffffff;
    cv[q] = -__builtin_inff(); ci[q] = 0x7fffffff;
  }
  for (int n = threadIdx.x; n < N_; n += TPB) {
    const size_t idx = ((size_t)(b * N_ + n)) * T_ + t;
    insert4(iv, ii, ws_iou[idx], n);
    insert4(cv, ci, -ws_cost[idx], n);
  }

  // ---- reduce iou top-4, compute dyn_k ----
#pragma unroll
  for (int q = 0; q < Q_; ++q) { sv[threadIdx.x * Q_ + q] = iv[q]; si[threadIdx.x * Q_ + q] = ii[q]; }
  __syncthreads();
  if (threadIdx.x == 0) {
    float bv[Q_]; int bi[Q_];
#pragma unroll
    for (int q = 0; q < Q_; ++q) { bv[q] = -__builtin_inff(); bi[q] = 0x7fffffff; }
    for (int k = 0; k < TPB * Q_; ++k) insert4(bv, bi, sv[k], si[k]);
    const float s = ((bv[0] + bv[1]) + bv[2]) + bv[3];
    const int valid = masks[b * T_ + t];
    int dk = (int)s;              // truncation, matches astype(int32) for s >= 0
    if (dk < 1) dk = 1;
    ws_dynk[b * T_ + t] = valid ? dk : 0;
  }
  __syncthreads();

  // ---- reduce (-cost) top-4 indices ----
#pragma unroll
  for (int q = 0; q < Q_; ++q) { sv[threadIdx.x * Q_ + q] = cv[q]; si[threadIdx.x * Q_ + q] = ci[q]; }
  __syncthreads();
  if (threadIdx.x == 0) {
    float bv[Q_]; int bi[Q_];
#pragma unroll
    for (int q = 0; q < Q_; ++q) { bv[q] = -__builtin_inff(); bi[q] = 0x7fffffff; }
    for (int k = 0; k < TPB * Q_; ++k) insert4(bv, bi, sv[k], si[k]);
#pragma unroll
    for (int q = 0; q < Q_; ++q) ws_topk[(b * T_ + t) * Q_ + q] = bi[q];
  }
}

// assign: membership count, conflict -> argmin(cost), write outputs as floats
__global__ __launch_bounds__(TPB) void k_assign(
    const float* __restrict__ ws_cost, const int* __restrict__ ws_dynk,
    const int* __restrict__ ws_topk, float* __restrict__ out) {
  __shared__ int s_dk[T_];
  __shared__ int s_ix[T_ * Q_];
  const int b = blockIdx.x / NTILES;
  const int n = (blockIdx.x % NTILES) * TPB + threadIdx.x;
  if (threadIdx.x < T_)      s_dk[threadIdx.x] = ws_dynk[b * T_ + threadIdx.x];
  if (threadIdx.x < T_ * Q_) s_ix[threadIdx.x] = ws_topk[b * T_ * Q_ + threadIdx.x];
  __syncthreads();

  int cnt = 0, first = -1;
  for (int t = 0; t < T_; ++t) {
    const int k = s_dk[t];
#pragma unroll
    for (int q = 0; q < Q_; ++q) {
      if (q < k && s_ix[t * Q_ + q] == n) { cnt++; if (first < 0) first = t; }
    }
  }

  int matched;
  if (cnt == 0) {
    matched = -1;
  } else if (cnt == 1) {
    matched = first;
  } else {
    const size_t row = ((size_t)(b * N_ + n)) * T_;
    float best = ws_cost[row];
    int bi = 0;
    for (int t = 1; t < T_; ++t) {
      const float c = ws_cost[row + t];
      if (c < best) { best = c; bi = t; }  // first occurrence of minimum
    }
    matched = bi;
  }
  out[(size_t)b * N_ + n]            = (cnt > 0) ? 1.f : 0.f;
  out[(size_t)B_ * N_ + b * N_ + n]  = (float)matched;
}

// ---------------------------------------------------------------------------
extern "C" void kernel_launch(void* const* d_in, const int* in_sizes, int n_in,
                              void* d_out, int out_size, void* d_ws, size_t ws_size,
                              hipStream_t stream) {
  (void)in_sizes; (void)n_in; (void)out_size; (void)ws_size;
  const float* preds = (const float*)d_in[0];
  const float* tgts  = (const float*)d_in[1];
  const int*   masks = (const int*)d_in[2];
  const int*   iwp   = (const int*)d_in[3];
  // d_in[4] = img_h (unused by reference)

  float* ws      = (float*)d_ws;
  float* ws_iou  = ws;                               // B*N*T floats
  float* ws_cost = ws + (size_t)B_ * N_ * T_;        // B*N*T floats
  unsigned int* ws_max = (unsigned int*)(ws + 2 * (size_t)B_ * N_ * T_); // B*3
  int* ws_dynk = (int*)(ws_max + B_ * 3);            // B*T
  int* ws_topk = ws_dynk + B_ * T_;                  // B*T*Q
  float* out = (float*)d_out;                        // [assigned(B*N), matched(B*N)]

  k_init  <<<1, 64, 0, stream>>>(ws_max);
  k_phase1<<<B_ * NTILES, TPB, 0, stream>>>(preds, tgts, masks, iwp, ws_iou, ws_max);
  k_phase2<<<B_ * NTILES, TPB, 0, stream>>>(preds, tgts, masks, iwp, ws_cost, ws_max);
  k_topk  <<<B_ * T_,     TPB, 0, stream>>>(ws_iou, ws_cost, masks, ws_dynk, ws_topk);
  k_assign<<<B_ * NTILES, TPB, 0, stream>>>(ws_cost, ws_dynk, ws_topk, out);
}